// DigitCaps_7327214207144
// MI455X (gfx1250) — compile-verified
//
#include <hip/hip_runtime.h>
#include <math.h>

#define B_  256
#define O_  10
#define I_  1152
#define D_  16
#define K_  8
#define EPSQ 1e-8f

typedef __attribute__((ext_vector_type(2))) float v2f;
typedef __attribute__((ext_vector_type(8))) float v8f;

// ---------------------------------------------------------------- zero logits
__global__ void k_zero(float* __restrict__ p, int n) {
    for (int idx = blockIdx.x * blockDim.x + threadIdx.x; idx < n;
         idx += gridDim.x * blockDim.x)
        p[idx] = 0.0f;
}

// ------------------------------------------------- c = softmax_o(logits[b,:,i])
__global__ void k_softmax_c(const float* __restrict__ logits,
                            float* __restrict__ cbuf) {
    int t = blockIdx.x * blockDim.x + threadIdx.x;   // one thread per (b,i)
    if (t >= B_ * I_) return;
    int b = t / I_, i = t % I_;
    const float* lp = logits + (size_t)b * O_ * I_ + i;
    float l[O_], m = -1e30f;
#pragma unroll
    for (int o = 0; o < O_; ++o) { l[o] = lp[o * I_]; m = fmaxf(m, l[o]); }
    float s = 0.0f;
#pragma unroll
    for (int o = 0; o < O_; ++o) { l[o] = __expf(l[o] - m); s += l[o]; }
    float inv = 1.0f / s;
    float* cp = cbuf + (size_t)b * O_ * I_ + i;
#pragma unroll
    for (int o = 0; o < O_; ++o) cp[o * I_] = l[o] * inv;
}

// --------------- s[b,o,:] = sum_{i,k} (c[b,o,i]*x[b,i,k]) * W[o,i,:,k]  (WMMA)
// One workgroup per (b-tile of 16, o). 8 waves split the K = I_*K_ = 9216
// contraction; partial 16x16 C tiles reduced through LDS.
__global__ __launch_bounds__(256) void k_s_pass(
        const float* __restrict__ x, const float* __restrict__ W,
        const float* __restrict__ cbuf, float* __restrict__ s_out) {
    const int b0   = blockIdx.x * 16;
    const int o    = blockIdx.y;
    const int wave = threadIdx.x >> 5;
    const int lane = threadIdx.x & 31;
    const int half = lane >> 4;       // K-pair selector for A/B operands
    const int lcol = lane & 15;       // A: row M (=batch), B: col N (=out_dim)

    v8f acc = {0.f, 0.f, 0.f, 0.f, 0.f, 0.f, 0.f, 0.f};

    const int chunks_per_wave = (I_ * K_ / 4) / 8;   // 2304/8 = 288
    const int c4beg = wave * chunks_per_wave;

    const size_t xrow  = (size_t)(b0 + lcol) * (I_ * K_);
    const size_t crow  = (size_t)(b0 + lcol) * (O_ * I_) + (size_t)o * I_;
    const size_t wbase = (size_t)o * I_ * (D_ * K_);

    for (int c4 = c4beg; c4 < c4beg + chunks_per_wave; ++c4) {
        const int kbase = c4 * 4;
        const int i  = kbase >> 3;    // input capsule
        const int k0 = kbase & 7;     // 0 or 4 inside that capsule
        const float cv = cbuf[crow + i];
        const float* xp = x + xrow + i * K_ + k0 + 2 * half;
        v2f a;  a.x = cv * xp[0];  a.y = cv * xp[1];
        const float* wp = W + wbase + ((size_t)i * D_ + lcol) * K_ + k0 + 2 * half;
        __builtin_prefetch(wp + D_ * K_, 0, 1);       // next capsule's W slice
        v2f bm; bm.x = wp[0];      bm.y = wp[1];
        acc = __builtin_amdgcn_wmma_f32_16x16x4_f32(
                  false, a, false, bm, (short)0, acc, false, false);
    }

    __shared__ float red[8][256];
#pragma unroll
    for (int r = 0; r < 8; ++r) red[wave][lane * 8 + r] = acc[r];
    __syncthreads();

    const int t = threadIdx.x;              // one thread per C element
    const int m = t >> 4, n = t & 15;
    const int r = m & 7, l = ((m >> 3) << 4) | n;
    float sum = 0.0f;
#pragma unroll
    for (int w = 0; w < 8; ++w) sum += red[w][l * 8 + r];
    s_out[((size_t)(b0 + m) * O_ + o) * D_ + n] = sum;
}

// ----- logits[b,o,i] += g(||s-u||^2) * (<u,s> - ||u||^2), u recomputed (WMMA)
__global__ __launch_bounds__(256) void k_update(
        const float* __restrict__ x, const float* __restrict__ W,
        const float* __restrict__ s_in, float* __restrict__ logits) {
    const int b0   = blockIdx.x * 16;
    const int o    = blockIdx.y;
    const int wave = threadIdx.x >> 5;
    const int lane = threadIdx.x & 31;
    const int half = lane >> 4;
    const int lcol = lane & 15;

    __shared__ float s_tile[16][16];
    __shared__ float snorm[16];
    {
        const int t = threadIdx.x;
        const int m = t >> 4, n = t & 15;
        s_tile[m][n] = s_in[((size_t)(b0 + m) * O_ + o) * D_ + n];
    }
    __syncthreads();
    if (threadIdx.x < 16) {
        float a = 0.0f;
#pragma unroll
        for (int n = 0; n < 16; ++n) { float v = s_tile[threadIdx.x][n]; a += v * v; }
        snorm[threadIdx.x] = a;
    }
    __syncthreads();

    const int ibase = blockIdx.z * 128 + wave * 16;   // 16 capsules per wave
    const size_t xrow  = (size_t)(b0 + lcol) * (I_ * K_);
    const size_t wbase = (size_t)o * I_ * (D_ * K_);

    for (int ii = 0; ii < 16; ++ii) {
        const int i = ibase + ii;
        v8f acc = {0.f, 0.f, 0.f, 0.f, 0.f, 0.f, 0.f, 0.f};
#pragma unroll
        for (int k0 = 0; k0 < K_; k0 += 4) {
            const float* xp = x + xrow + i * K_ + k0 + 2 * half;
            v2f a;  a.x = xp[0];  a.y = xp[1];
            const float* wp = W + wbase + ((size_t)i * D_ + lcol) * K_ + k0 + 2 * half;
            v2f bm; bm.x = wp[0]; bm.y = wp[1];
            acc = __builtin_amdgcn_wmma_f32_16x16x4_f32(
                      false, a, false, bm, (short)0, acc, false, false);
        }
        // acc holds u[b0+m, d]: m = r + 8*half across r=0..7, d = lcol.
        float pa[8], pn[8];
#pragma unroll
        for (int r = 0; r < 8; ++r) {
            const int m = r + 8 * half;
            const float u = acc[r];
            pa[r] = u * s_tile[m][lcol];   // -> <u, s>
            pn[r] = u * u;                 // -> ||u||^2
        }
#pragma unroll
        for (int mask = 1; mask < 16; mask <<= 1) {
#pragma unroll
            for (int r = 0; r < 8; ++r) {
                pa[r] += __shfl_xor(pa[r], mask, 32);
                pn[r] += __shfl_xor(pn[r], mask, 32);
            }
        }
        if (lcol == 0) {                   // lanes 0 and 16 own 8 rows each
#pragma unroll
            for (int r = 0; r < 8; ++r) {
                const int m = r + 8 * half;
                const float a = pa[r], nn = pn[r];
                float sq = snorm[m] - 2.0f * a + nn;
                sq = fmaxf(sq, 0.0f);
                const float g = sq / ((1.0f + sq) * (sqrtf(sq) + EPSQ));
                const size_t idx =
                    (size_t)(b0 + m) * (O_ * I_) + (size_t)o * I_ + i;
                logits[idx] += g * (a - nn);
            }
        }
    }
}

// ------------------------------------------------------- out = squash(s[b,o,:])
__global__ void k_out(const float* __restrict__ s_in, float* __restrict__ out) {
    int t = blockIdx.x * blockDim.x + threadIdx.x;   // one thread per (b,o)
    if (t >= B_ * O_) return;
    const float* sp = s_in + (size_t)t * D_;
    float sq = 0.0f;
#pragma unroll
    for (int d = 0; d < D_; ++d) sq += sp[d] * sp[d];
    const float g = sq / ((1.0f + sq) * (sqrtf(sq) + EPSQ));
    float* op = out + (size_t)t * D_;
#pragma unroll
    for (int d = 0; d < D_; ++d) op[d] = g * sp[d];
}

extern "C" void kernel_launch(void* const* d_in, const int* in_sizes, int n_in,
                              void* d_out, int out_size, void* d_ws, size_t ws_size,
                              hipStream_t stream) {
    const float* x = (const float*)d_in[0];
    // d_in[1] (y) is unused by the reference computation.
    const float* W = (const float*)d_in[2];
    float* out = (float*)d_out;
    float* ws  = (float*)d_ws;

    const size_t BOI = (size_t)B_ * O_ * I_;           // 2,949,120
    float* c_buf  = ws;                                 // 11.8 MB
    float* logits = ws + BOI;                           // 11.8 MB
    float* s_buf  = ws + 2 * BOI;                       // 160 KB

    hipLaunchKernelGGL(k_zero, dim3(512), dim3(256), 0, stream,
                       logits, (int)BOI);

    for (int t = 0; t < 3; ++t) {                       // NUM_ROUTING = 3
        hipLaunchKernelGGL(k_softmax_c, dim3((B_ * I_ + 255) / 256), dim3(256),
                           0, stream, logits, c_buf);
        hipLaunchKernelGGL(k_s_pass, dim3(B_ / 16, O_), dim3(256),
                           0, stream, x, W, c_buf, s_buf);
        if (t < 2)
            hipLaunchKernelGGL(k_update, dim3(B_ / 16, O_, I_ / 128), dim3(256),
                               0, stream, x, W, s_buf, logits);
    }
    hipLaunchKernelGGL(k_out, dim3((B_ * O_ + 255) / 256), dim3(256),
                       0, stream, s_buf, out);
}